// RationalsModel_53369263620373
// MI455X (gfx1250) — compile-verified
//
#include <hip/hip_runtime.h>

typedef __attribute__((ext_vector_type(2))) float v2f;
typedef __attribute__((ext_vector_type(8))) float v8f;

// Swap lane c <-> lane c+16 (wave32) in one LDS-pipe op (SWAPX16).
__device__ __forceinline__ float swap16(float v)
{
    int i = __builtin_bit_cast(int, v);
    i = __builtin_amdgcn_ds_swizzle(i, 0x401F);
    return __builtin_bit_cast(float, i);
}

// Fast reciprocal with one Newton refinement (t >= 1 here, well conditioned).
__device__ __forceinline__ float rcp_nr(float t)
{
    float r = __builtin_amdgcn_rcpf(t);
    r = r * fmaf(-t, r, 2.0f);
    return r;
}

// One 16-element tile. A-matrix rows 0-7 = numerator coeff chunks,
// rows 8-15 = denominator coeff chunks, so D hands lane c the 8 num
// partials and lane c+16 the 8 den partials -> one Horner chain per lane.
// Returns, in EVERY lane, out = num/(|den|+1) for element (base + (lane&15)).
__device__ __forceinline__ float tile_eval(float xv, bool hiHalf, v2f A1, v2f A2)
{
    const float x2 = xv * xv;
    const float x3 = x2 * xv;
    const float x4 = x2 * x2;

    // B1 4x16: rows x^0..x^3 (VGPR0=row (hi?2:0), VGPR1=row (hi?3:1))
    v2f B1, B2;
    B1.x = hiHalf ? x2 : 1.0f;
    B1.y = hiHalf ? x3 : xv;
    // B2 rows x^4..x^7 = x^4 * B1 rows
    B2.x = x4 * B1.x;
    B2.y = x4 * B1.y;

    v8f cz = {0.f, 0.f, 0.f, 0.f, 0.f, 0.f, 0.f, 0.f};

    // Chained accumulate: D[r][c] = sum_{k=0..7} coef_chunk[r][k] * x_c^k
    v8f D = __builtin_amdgcn_wmma_f32_16x16x4_f32(
        false, A1, false, B1, (short)0, cz, false, false);
    D = __builtin_amdgcn_wmma_f32_16x16x4_f32(
        false, A2, false, B2, (short)0, D, false, false);

    const float y = x4 * x4;  // x^8

    // One 8-term Horner per lane: lo lanes -> num poly, hi lanes -> den poly.
    float S = D[7];
#pragma unroll
    for (int r = 6; r >= 0; --r) S = fmaf(S, y, D[r]);

    // Denominator side: den = x * dpoly, t = |den| + 1 (abs is free modifier).
    float t = fmaf(fabsf(xv * S), 1.0f, 1.0f);  // |x*S| + 1
    float sendv = hiHalf ? t : S;               // hi sends processed den
    float recv  = swap16(sendv);                // lo receives t, hi receives num

    float numv = hiHalf ? recv : S;
    float tv   = hiHalf ? t    : recv;
    return numv * rcp_nr(tv);
}

__global__ __launch_bounds__(256)
void rationals_wmma_kernel(const float* __restrict__ x,
                           const float* __restrict__ coef,
                           float* __restrict__ out, int N)
{
    const int lane = threadIdx.x & 31;
    const int col  = lane & 15;
    const bool hi  = (lane >= 16);

    // A row = col. Rows 0-7: num chunks coef[8*row .. 8*row+7];
    // rows 8-15: den chunks coef[64 + 8*(row-8) ..] == coef[8*row ..]. So the
    // concatenated coefficient array maps directly: chunk base = 8*col.
    const int kb = hi ? 2 : 0;              // K-split across lane halves
    const int cb = 8 * col;
    v2f A1, A2;
    A1.x = coef[cb + kb + 0];
    A1.y = coef[cb + kb + 1];
    A2.x = coef[cb + 4 + kb + 0];
    A2.y = coef[cb + 4 + kb + 1];

    const int wid        = (blockIdx.x * blockDim.x + threadIdx.x) >> 5; // wave-uniform
    const int totalWaves = (gridDim.x * blockDim.x) >> 5;

    for (int base = wid * 64; base < N; base += totalWaves * 64) {
        if (base + 64 <= N) {
            // Four independent 16-element tiles -> 64 consecutive elements.
            float x0 = x[base + col];
            float x1 = x[base + 16 + col];
            float x2 = x[base + 32 + col];
            float x3 = x[base + 48 + col];

            float r0 = tile_eval(x0, hi, A1, A2);
            float r1 = tile_eval(x1, hi, A1, A2);
            float r2 = tile_eval(x2, hi, A1, A2);
            float r3 = tile_eval(x3, hi, A1, A2);

            // lane l stores elements base+l and base+32+l (coalesced b32 x2).
            out[base + lane]      = hi ? r1 : r0;
            out[base + 32 + lane] = hi ? r3 : r2;
        } else {
            // Scalar tail (no WMMA under partial EXEC).
#pragma unroll
            for (int e = 0; e < 64; e += 32) {
                int idx = base + e + lane;
                if (idx < N) {
                    float xv = x[idx];
                    float sn = coef[63];
                    float sd = coef[127];
                    for (int p = 62; p >= 0; --p) {
                        sn = fmaf(sn, xv, coef[p]);
                        sd = fmaf(sd, xv, coef[64 + p]);
                    }
                    sd *= xv;
                    out[idx] = sn / (fabsf(sd) + 1.0f);
                }
            }
        }
    }
}

extern "C" void kernel_launch(void* const* d_in, const int* in_sizes, int n_in,
                              void* d_out, int out_size, void* d_ws, size_t ws_size,
                              hipStream_t stream)
{
    const float* x    = (const float*)d_in[0];
    const float* coef = (const float*)d_in[1];
    float* out        = (float*)d_out;
    const int N       = in_sizes[0];

    const int block = 256;                   // 8 waves; 64 elements/wave/iter
    int grid = (N + 511) / 512;              // 512 elements per block per iter
    if (grid > 4096) grid = 4096;            // grid-stride loop covers the rest
    if (grid < 1) grid = 1;

    rationals_wmma_kernel<<<dim3(grid), dim3(block), 0, stream>>>(x, coef, out, N);
}